// SSM_15573551416046
// MI455X (gfx1250) — compile-verified
//
#include <hip/hip_runtime.h>
#include <hip/hip_bf16.h>
#include <math.h>

typedef __attribute__((ext_vector_type(2))) float v2f;
typedef __attribute__((ext_vector_type(8))) float v8f;

// D = A(16x4) * B(4x16) + C(16x16), all f32, wave32.
#define WMMA4(a, b, c) \
  __builtin_amdgcn_wmma_f32_16x16x4_f32(false, (a), false, (b), (short)0, (c), false, false)

static constexpr int Bn = 8;
static constexpr int Ln = 2048;
static constexpr int Dn = 2048;
static constexpr int NK = Dn / 4;         // 512 K-steps of 4
static constexpr int CH = 128;            // scan chunk length
static constexpr int NCH = Ln / CH;       // 16 chunks per batch

__device__ __forceinline__ void load_row16(const float* __restrict__ p, float* rr) {
  float4 a = *(const float4*)(p + 0);
  float4 b = *(const float4*)(p + 4);
  float4 c = *(const float4*)(p + 8);
  float4 d = *(const float4*)(p + 12);
  rr[0]=a.x; rr[1]=a.y; rr[2]=a.z; rr[3]=a.w;
  rr[4]=b.x; rr[5]=b.y; rr[6]=b.z; rr[7]=b.w;
  rr[8]=c.x; rr[9]=c.y; rr[10]=c.z; rr[11]=c.w;
  rr[12]=d.x; rr[13]=d.y; rr[14]=d.z; rr[15]=d.w;
}

__device__ __forceinline__ float matvec16(const float* rr, float v) {
  float acc = 0.f;
#pragma unroll
  for (int j = 0; j < 16; ++j) acc = fmaf(rr[j], __shfl(v, j, 32), acc);
  return acc;
}

// ---------------------------------------------------------------------------
// Kernel 0: pre-fragment the weight matrices into per-lane WMMA B-fragment
// order so the projection hot loop is branch-free and fully coalesced.
// For step s, lane l: kk = 4*s + 2*(l>>4), n = l&15 ->
//   F[(s*32+l)*2 + {0,1}] = { W^T[kk][n], W^T[kk+1][n] }
// ---------------------------------------------------------------------------
__global__ __launch_bounds__(256) void frag_weights_kernel(
    const float* __restrict__ dp, const float* __restrict__ Wb,
    const float* __restrict__ Wc, const float* __restrict__ Wd,
    float* __restrict__ WbF, float* __restrict__ WcF, float* __restrict__ WxF) {
  const int idx = blockIdx.x * 256 + threadIdx.x;   // [0, NK*32)
  const int lane = idx & 31;
  const int s = idx >> 5;
  const int hi = lane >> 4;
  const int nm = lane & 15;
  const int kk = 4 * s + 2 * hi;

  WbF[idx * 2 + 0] = Wb[(size_t)nm * Dn + kk];
  WbF[idx * 2 + 1] = Wb[(size_t)nm * Dn + kk + 1];
  WcF[idx * 2 + 0] = Wc[(size_t)nm * Dn + kk];
  WcF[idx * 2 + 1] = Wc[(size_t)nm * Dn + kk + 1];
  WxF[idx * 2 + 0] = (nm == 0) ? dp[kk]     : ((nm == 1) ? Wd[kk]     : 0.f);
  WxF[idx * 2 + 1] = (nm == 0) ? dp[kk + 1] : ((nm == 1) ? Wd[kk + 1] : 0.f);
}

// ---------------------------------------------------------------------------
// Kernel 1: delta / Bt / Ct / Dt projections. One wave per 16-row tile of u.
// u rows -> WMMA A-frag (16x4 f32), pre-fragmented weights -> B-frags.
// ---------------------------------------------------------------------------
__global__ __launch_bounds__(256) void proj_kernel(
    const float* __restrict__ u,
    const float* __restrict__ WbF, const float* __restrict__ WcF,
    const float* __restrict__ WxF,
    const float* __restrict__ bb, const float* __restrict__ bc,
    const float* __restrict__ bd,
    float* __restrict__ delta, float* __restrict__ Bt,
    float* __restrict__ Ct, float* __restrict__ Dt) {
  const int lane = threadIdx.x & 31;
  const int wave = threadIdx.x >> 5;
  const int tile = blockIdx.x * 8 + wave;   // 1024 tiles over B*L rows
  const int row0 = tile * 16;
  const int hi = lane >> 4;
  const int nm = lane & 15;                 // M for A-frag, N for B-frag

  const float* urow = u + (size_t)(row0 + nm) * Dn + 2 * hi;
  const v2f* wbp = (const v2f*)WbF + lane;
  const v2f* wcp = (const v2f*)WcF + lane;
  const v2f* wxp = (const v2f*)WxF + lane;

  v8f accB = {}; v8f accC = {}; v8f accX = {};
  for (int s = 0; s < NK; ++s) {
    v2f a  = *(const v2f*)(urow + 4 * s);
    v2f wb = wbp[s * 32];
    v2f wc = wcp[s * 32];
    v2f wx = wxp[s * 32];
    accB = WMMA4(a, wb, accB);
    accC = WMMA4(a, wc, accC);
    accX = WMMA4(a, wx, accX);
  }
#pragma unroll
  for (int v = 0; v < 8; ++v) {
    const int row = row0 + v + 8 * hi;
    Bt[(size_t)row * 16 + nm] = accB[v] + bb[nm];
    Ct[(size_t)row * 16 + nm] = accC[v] + bc[nm];
    if (nm == 0) delta[row] = 1.f / (1.f + expf(-accX[v]));
    if (nm == 1) Dt[row] = accX[v] + bd[0];
  }
}

// ---------------------------------------------------------------------------
// Kernel 2: Ad[b,t] = expm(A[t] * delta[b,t]).  One wave per (b,t).
// Scaling-and-squaring Taylor (order 10, 6 squarings), 64 WMMAs per wave.
// C-layout <-> A/B-frag conversion through a private 1KB LDS tile.
// ---------------------------------------------------------------------------
__global__ __launch_bounds__(256) void expm_kernel(
    const float* __restrict__ A, const float* __restrict__ delta,
    float* __restrict__ Ad) {
  __shared__ float lds[8][256];
  const int lane = threadIdx.x & 31;
  const int wave = threadIdx.x >> 5;
  const int idx = blockIdx.x * 8 + wave;    // b*L + t in [0, 16384)
  const int t = idx & (Ln - 1);
  const int hi = lane >> 4;
  const int nm = lane & 15;
  float* S = lds[wave];

  const float sc = delta[idx] * 0.015625f;  // 2^-6 pre-scale
  const float* Ab = A + (size_t)t * 256;

  // Right operand Ms in B-frag registers (constant across Taylor steps)
  v2f bM[4];
#pragma unroll
  for (int c = 0; c < 4; ++c) {
    const int kk = 4 * c + 2 * hi;
    v2f b;
    b.x = Ab[kk * 16 + nm] * sc;
    b.y = Ab[(kk + 1) * 16 + nm] * sc;
    bM[c] = b;
  }

  v8f out, term;
#pragma unroll
  for (int v = 0; v < 8; ++v) {
    const float e = ((v + 8 * hi) == nm) ? 1.f : 0.f;
    out[v] = e; term[v] = e;
  }

  constexpr float rk[11] = {0.f, 1.f, 0.5f, 1.f/3.f, 0.25f, 0.2f,
                            1.f/6.f, 1.f/7.f, 0.125f, 1.f/9.f, 0.1f};
#pragma unroll
  for (int k = 1; k <= 10; ++k) {
#pragma unroll
    for (int v = 0; v < 8; ++v) S[(v + 8 * hi) * 16 + nm] = term[v];
    v8f acc = {};
#pragma unroll
    for (int c = 0; c < 4; ++c) {
      const int kk = 4 * c + 2 * hi;
      v2f a = *(const v2f*)(&S[nm * 16 + kk]);
      acc = WMMA4(a, bM[c], acc);
    }
#pragma unroll
    for (int v = 0; v < 8; ++v) { term[v] = acc[v] * rk[k]; out[v] += term[v]; }
  }

#pragma unroll
  for (int s = 0; s < 6; ++s) {
#pragma unroll
    for (int v = 0; v < 8; ++v) S[(v + 8 * hi) * 16 + nm] = out[v];
    v8f acc = {};
#pragma unroll
    for (int c = 0; c < 4; ++c) {
      const int kk = 4 * c + 2 * hi;
      v2f a = *(const v2f*)(&S[nm * 16 + kk]);
      v2f b;
      b.x = S[kk * 16 + nm];
      b.y = S[(kk + 1) * 16 + nm];
      acc = WMMA4(a, b, acc);
    }
    out = acc;
  }

  float* Adp = Ad + (size_t)idx * 256;
#pragma unroll
  for (int v = 0; v < 8; ++v) Adp[(v + 8 * hi) * 16 + nm] = out[v];
}

// ---------------------------------------------------------------------------
// Kernel 3a: per-chunk aggregates. P_c = Ad[last]...Ad[first] (WMMA chain),
// q_c = local scan from zero (shuffle matvec). One wave per chunk.
// ---------------------------------------------------------------------------
__global__ __launch_bounds__(256) void chunk_kernel(
    const float* __restrict__ Ad, const float* __restrict__ Bt,
    float* __restrict__ PWS, float* __restrict__ qWS) {
  __shared__ float lds[8][256];
  const int lane = threadIdx.x & 31;
  const int wave = threadIdx.x >> 5;
  const int g = blockIdx.x * 8 + wave;      // [0, 128)
  const int b = g >> 4;
  const int c = g & (NCH - 1);
  const int t0 = c * CH;
  const int hi = lane >> 4;
  const int nm = lane & 15;
  float* S = lds[wave];

  v8f P;
#pragma unroll
  for (int v = 0; v < 8; ++v) P[v] = ((v + 8 * hi) == nm) ? 1.f : 0.f;
  float q = 0.f;

  for (int tt = 0; tt < CH; ++tt) {
    const size_t row = (size_t)(b * Ln + t0 + tt);
    const float* M = Ad + row * 256;
    float rr[16];
    load_row16(M + nm * 16, rr);
    const float nq = matvec16(rr, q);

#pragma unroll
    for (int v = 0; v < 8; ++v) S[(v + 8 * hi) * 16 + nm] = P[v];
    v8f acc = {};
#pragma unroll
    for (int c4 = 0; c4 < 4; ++c4) {
      const int kk = 4 * c4 + 2 * hi;
      v2f a = *(const v2f*)(&M[nm * 16 + kk]);   // A-frag of Ad[t] from global
      v2f bf;
      bf.x = S[kk * 16 + nm];                    // B-frag of P from LDS
      bf.y = S[(kk + 1) * 16 + nm];
      acc = WMMA4(a, bf, acc);
    }
    P = acc;
    q = nq + Bt[row * 16 + nm];
  }

  float* Pp = PWS + (size_t)g * 256;
#pragma unroll
  for (int v = 0; v < 8; ++v) Pp[(v + 8 * hi) * 16 + nm] = P[v];
  if (lane < 16) qWS[g * 16 + lane] = q;
}

// ---------------------------------------------------------------------------
// Kernel 3b: sequential carry over 16 chunks per batch (one wave per batch).
// Stores each chunk's entry state e_c; e_{c+1} = P_c @ e_c + q_c.
// ---------------------------------------------------------------------------
__global__ __launch_bounds__(256) void carry_kernel(
    const float* __restrict__ PWS, const float* __restrict__ qWS,
    float* __restrict__ EWS) {
  const int lane = threadIdx.x & 31;
  const int b = threadIdx.x >> 5;           // 8 waves = 8 batches
  const int nm = lane & 15;
  float e = 0.f;
  for (int c = 0; c < NCH; ++c) {
    const int g = b * NCH + c;
    if (lane < 16) EWS[g * 16 + lane] = e;
    float rr[16];
    load_row16(PWS + (size_t)g * 256 + nm * 16, rr);
    const float ne = matvec16(rr, e);
    e = ne + qWS[g * 16 + nm];
  }
}

// ---------------------------------------------------------------------------
// Kernel 3c: re-scan each chunk from its entry state, emit y = Ct.s + Dt.
// ---------------------------------------------------------------------------
__global__ __launch_bounds__(256) void scan_kernel(
    const float* __restrict__ Ad, const float* __restrict__ Bt,
    const float* __restrict__ Ct, const float* __restrict__ Dt,
    const float* __restrict__ EWS, float* __restrict__ y) {
  const int lane = threadIdx.x & 31;
  const int wave = threadIdx.x >> 5;
  const int g = blockIdx.x * 8 + wave;
  const int b = g >> 4;
  const int c = g & (NCH - 1);
  const int t0 = c * CH;
  const int nm = lane & 15;

  float s = EWS[g * 16 + nm];
  for (int tt = 0; tt < CH; ++tt) {
    const size_t row = (size_t)(b * Ln + t0 + tt);
    float rr[16];
    load_row16(Ad + row * 256 + nm * 16, rr);
    const float ns = matvec16(rr, s);
    s = ns + Bt[row * 16 + nm];
    float p = Ct[row * 16 + nm] * s;
    p += __shfl_xor(p, 8, 32);
    p += __shfl_xor(p, 4, 32);
    p += __shfl_xor(p, 2, 32);
    p += __shfl_xor(p, 1, 32);
    if (lane == 0) y[row] = p + Dt[row];
  }
}

// ---------------------------------------------------------------------------
extern "C" void kernel_launch(void* const* d_in, const int* in_sizes, int n_in,
                              void* d_out, int out_size, void* d_ws, size_t ws_size,
                              hipStream_t stream) {
  const float* u   = (const float*)d_in[0];
  const float* A   = (const float*)d_in[1];
  const float* dp  = (const float*)d_in[2];
  const float* Wb  = (const float*)d_in[3];
  const float* bb  = (const float*)d_in[4];
  const float* Wc  = (const float*)d_in[5];
  const float* bc  = (const float*)d_in[6];
  const float* Wd  = (const float*)d_in[7];
  const float* bd  = (const float*)d_in[8];
  float* out = (float*)d_out;

  float* ws    = (float*)d_ws;
  float* delta = ws;                         // B*L            = 16384
  float* Bt    = delta + Bn * Ln;            // B*L*16         = 262144
  float* Ct    = Bt + Bn * Ln * 16;          // 262144
  float* Dt    = Ct + Bn * Ln * 16;          // 16384
  float* Ad    = Dt + Bn * Ln;               // B*L*256        = 4194304
  float* PWS   = Ad + (size_t)Bn * Ln * 256; // 128*256        = 32768
  float* qWS   = PWS + Bn * NCH * 256;       // 128*16         = 2048
  float* EWS   = qWS + Bn * NCH * 16;        // 2048
  float* WbF   = EWS + Bn * NCH * 16;        // NK*64          = 32768
  float* WcF   = WbF + NK * 64;              // 32768
  float* WxF   = WcF + NK * 64;              // 32768

  frag_weights_kernel<<<64, 256, 0, stream>>>(dp, Wb, Wc, Wd, WbF, WcF, WxF);
  proj_kernel<<<128, 256, 0, stream>>>(u, WbF, WcF, WxF, bb, bc, bd,
                                       delta, Bt, Ct, Dt);
  expm_kernel<<<2048, 256, 0, stream>>>(A, delta, Ad);
  chunk_kernel<<<16, 256, 0, stream>>>(Ad, Bt, PWS, qWS);
  carry_kernel<<<1, 256, 0, stream>>>(PWS, qWS, EWS);
  scan_kernel<<<16, 256, 0, stream>>>(Ad, Bt, Ct, Dt, EWS, out);
}